// differentiableSuperpixelEmbedding_67577015435408
// MI455X (gfx1250) — compile-verified
//
#include <hip/hip_runtime.h>
#include <hip/hip_bf16.h>

#define B_ 8
#define C_ 3
#define N_ (224*224)   /* 50176 */
#define S_ 196
#define E_ 768
#define PIX 1024       /* pixels per block in accumulation kernel */

typedef float v2f __attribute__((ext_vector_type(2)));
typedef float v8f __attribute__((ext_vector_type(8)));

// ---------------------------------------------------------------------------
// Kernel 0: zero the global per-(b,s) accumulator [B][S][5]
// ---------------------------------------------------------------------------
__global__ void k_zero(float* __restrict__ p, int n) {
    int i = blockIdx.x * blockDim.x + threadIdx.x;
    if (i < n) p[i] = 0.0f;
}

// ---------------------------------------------------------------------------
// Kernel 1: per-pixel argmax labeling + exp-score accumulation.
// gaccum[b][s] = { sumExp, sumExp*x0, sumExp*x1, sumExp*x2, count }
// ---------------------------------------------------------------------------
__global__ void k_accum(const float* __restrict__ img,
                        const float* __restrict__ W_seg,
                        const float* __restrict__ b_seg,
                        const float* __restrict__ w_q,
                        float* __restrict__ gaccum) {
    __shared__ float ws0[S_], ws1[S_], ws2[S_], bs[S_];
    __shared__ float acc[S_ * 5];

    const int tid = threadIdx.x;
    const int b   = blockIdx.y;

    for (int s = tid; s < S_; s += blockDim.x) {
        ws0[s] = W_seg[s * 3 + 0];
        ws1[s] = W_seg[s * 3 + 1];
        ws2[s] = W_seg[s * 3 + 2];
        bs[s]  = b_seg[s];
    }
    for (int i = tid; i < S_ * 5; i += blockDim.x) acc[i] = 0.0f;
    __syncthreads();

    const float wq0 = w_q[0], wq1 = w_q[1], wq2 = w_q[2];
    const float* im = img + (size_t)b * (C_ * N_);
    const int base = blockIdx.x * PIX;

    #pragma unroll
    for (int i = 0; i < PIX / 256; ++i) {
        const int n = base + i * 256 + tid;
        if (n < N_) {
            const float x0 = im[n];
            const float x1 = im[N_ + n];
            const float x2 = im[2 * N_ + n];

            // argmax over S segment logits (strict '>' => first-max, as jnp.argmax)
            float best = -3.402823466e38f;
            int   bi   = 0;
            #pragma unroll 4
            for (int s = 0; s < S_; ++s) {
                float v = fmaf(ws0[s], x0, fmaf(ws1[s], x1, fmaf(ws2[s], x2, bs[s])));
                if (v > best) { best = v; bi = s; }
            }

            const float sc = fmaf(wq0, x0, fmaf(wq1, x1, wq2 * x2));
            const float e  = __expf(sc);

            float* a = &acc[bi * 5];
            atomicAdd(&a[0], e);
            atomicAdd(&a[1], e * x0);
            atomicAdd(&a[2], e * x1);
            atomicAdd(&a[3], e * x2);
            atomicAdd(&a[4], 1.0f);
        }
    }
    __syncthreads();

    float* gout = gaccum + (size_t)b * (S_ * 5);
    for (int i = tid; i < S_ * 5; i += blockDim.x) {
        const float v = acc[i];
        if (v != 0.0f) atomicAdd(&gout[i], v);
    }
}

// ---------------------------------------------------------------------------
// Kernel 2: finalize pooled vectors and run the (BS x 4) x (4 x E) GEMM with
// V_WMMA_F32_16X16X4_F32. One wave32 per 16x16 output tile; b_out (masked by
// segment presence) is folded into the C accumulator.
// ---------------------------------------------------------------------------
__global__ void k_wmma(const float* __restrict__ gaccum,
                       const float* __restrict__ W_out,
                       const float* __restrict__ b_out,
                       float* __restrict__ out) {
    const int TILES_N = E_ / 16;                                   // 48
    const int wave = (blockIdx.x * blockDim.x + threadIdx.x) >> 5; // global wave id
    const int lane = threadIdx.x & 31;
    const int tileM = wave / TILES_N;                              // 0..97
    const int tileN = wave % TILES_N;                              // 0..47
    const int m0 = tileM * 16;
    const int n0 = tileN * 16;

    const int  r  = lane & 15;
    const bool hi = (lane >= 16);

    // ---- A matrix: pooled[m0+r][k]; lanes 0-15 hold K=0,1; lanes 16-31 K=2,3
    const float* a = gaccum + (size_t)(m0 + r) * 5;
    const float sumE = a[0];
    const float sx0  = a[1];
    const float sx1  = a[2];
    const float sx2  = a[3];
    const float cnt  = a[4];
    const float denom = sumE + ((float)N_ - cnt);  // softmax denominator
    const float inv   = 1.0f / denom;              // safe: denom >= 1 always
    const float p0 = sx0 * inv;
    const float p1 = sx1 * inv;
    const float p2 = sx2 * inv;
    const float presentF = (cnt > 0.5f) ? 1.0f : 0.0f;

    v2f A;
    A.x = hi ? p2 : p0;     // K=2 : K=0
    A.y = hi ? 0.0f : p1;   // K=3(pad) : K=1

    // ---- B matrix: B[k][n] = W_out[(n0+n)*3 + k], K=3 row is zero padding
    const int e = n0 + r;   // output/embedding column for this lane
    v2f Bm;
    Bm.x = hi ? W_out[e * 3 + 2] : W_out[e * 3 + 0];
    Bm.y = hi ? 0.0f            : W_out[e * 3 + 1];

    // ---- C matrix: C[j] -> element (M = j + 8*hi, N = r). Fold in bias,
    // masked by presence of row m0 + j + 8*hi (flag held by lane j+8*hi).
    const float bo = b_out[e];
    v8f Cm;
    #pragma unroll
    for (int j = 0; j < 8; ++j) {
        const float pf = __shfl(presentF, j + (hi ? 8 : 0), 32);
        Cm[j] = pf * bo;
    }

    // ---- D = A x B + C   (v_wmma_f32_16x16x4_f32)
    v8f D = __builtin_amdgcn_wmma_f32_16x16x4_f32(
        /*neg_a=*/false, A, /*neg_b=*/false, Bm,
        /*c_mod=*/(short)0, Cm, /*reuse_a=*/false, /*reuse_b=*/false);

    // ---- store: VGPR j, this lane -> out[m0 + j + 8*hi][e]; 16 consecutive
    // floats per half-wave per j => coalesced 64B segments.
    #pragma unroll
    for (int j = 0; j < 8; ++j) {
        const int row = m0 + j + (hi ? 8 : 0);
        out[(size_t)row * E_ + e] = D[j];
    }
}

// ---------------------------------------------------------------------------
extern "C" void kernel_launch(void* const* d_in, const int* in_sizes, int n_in,
                              void* d_out, int out_size, void* d_ws, size_t ws_size,
                              hipStream_t stream) {
    (void)in_sizes; (void)n_in; (void)out_size; (void)ws_size;

    const float* img   = (const float*)d_in[0];  // [B,C,H,W]
    const float* W_seg = (const float*)d_in[1];  // [S,C]
    const float* b_seg = (const float*)d_in[2];  // [S]
    const float* w_q   = (const float*)d_in[3];  // [C]
    const float* W_out = (const float*)d_in[4];  // [E,C]
    const float* b_out = (const float*)d_in[5];  // [E]
    float*       out   = (float*)d_out;          // [B,S,E]
    float*       gacc  = (float*)d_ws;           // [B,S,5]

    const int accN = B_ * S_ * 5;                // 7840
    k_zero<<<(accN + 255) / 256, 256, 0, stream>>>(gacc, accN);

    dim3 g1((N_ + PIX - 1) / PIX, B_);           // 49 x 8
    k_accum<<<g1, 256, 0, stream>>>(img, W_seg, b_seg, w_q, gacc);

    // (B*S/16) * (E/16) = 98*48 = 4704 waves; 8 waves per 256-thread block
    const int totalWaves = (B_ * S_ / 16) * (E_ / 16);
    k_wmma<<<totalWaves / 8, 256, 0, stream>>>(gacc, W_out, b_out, out);
}